// NSVQ_33457795236535
// MI455X (gfx1250) — compile-verified
//
#include <hip/hip_runtime.h>
#include <hip/hip_bf16.h>
#include <math.h>

// ---------- types ----------
typedef __attribute__((ext_vector_type(2))) float v2f;
typedef __attribute__((ext_vector_type(8))) float v8f;

__device__ __forceinline__ v8f wmma4(v2f a, v2f b, v8f c) {
    // D = A(16x4,f32) * B(4x16,f32) + C(16x16,f32)
    return __builtin_amdgcn_wmma_f32_16x16x4_f32(
        false, a, false, b, (short)0, c, false, false);
}

// Problem constants
#define Bsz   128
#define Ssz   256
#define DIMsz 512
#define Dsz   128
#define Ksz   16384
#define Nrow  8192          // B*64
#define NEPS  1e-12f

// ===================================================================
// 1) h = (last - first) @ W_in     [32768,512]x[512,128] -> [32768,128]
//    (b_in cancels in the later subtraction; encoder is affine)
// ===================================================================
__global__ void encode_gemm(const float* __restrict__ first,
                            const float* __restrict__ last,
                            const float* __restrict__ Win,
                            float* __restrict__ h) {
    const int w = blockIdx.x * 8 + (threadIdx.x >> 5);
    const int rowTile = w >> 3, colTile = w & 7;          // 2048 x 8 tiles
    const int l = threadIdx.x & 31, half = l >> 4, q = l & 15;
    const int arow = rowTile * 16 + q;
    const int col  = colTile * 16 + q;
    v8f acc = {};
    #pragma unroll 8
    for (int k0 = 0; k0 < DIMsz; k0 += 4) {
        const int kk = k0 + 2 * half;
        float2 lf = *(const float2*)(last  + (size_t)arow * DIMsz + kk);
        float2 ff = *(const float2*)(first + (size_t)arow * DIMsz + kk);
        v2f a, b;
        a.x = lf.x - ff.x;  a.y = lf.y - ff.y;
        b.x = Win[(size_t)kk * Dsz + col];
        b.y = Win[(size_t)(kk + 1) * Dsz + col];
        acc = wmma4(a, b, acc);
    }
    const int r0 = rowTile * 16 + 8 * half;
    #pragma unroll
    for (int j = 0; j < 8; ++j)
        h[(size_t)(r0 + j) * Dsz + col] = acc[j];
}

// ===================================================================
// 2) x = conv3x3_s2(h as [B,D,16,16])  ->  [8192,128]
//    implicit im2col: A[m,(kh,kw,c)] = h[b, ih*16+iw, c] (contig in c)
// ===================================================================
__global__ void conv_gemm(const float* __restrict__ h,
                          const float* __restrict__ cw,   // [O=128,I=128,3,3]
                          float* __restrict__ x) {
    const int w = blockIdx.x * 8 + (threadIdx.x >> 5);
    const int rowTile = w >> 3, colTile = w & 7;          // 512 x 8 tiles
    const int l = threadIdx.x & 31, half = l >> 4, q = l & 15;
    const int m = rowTile * 16 + q;
    const int bimg = m >> 6, p = m & 63, oh = p >> 3, ow = p & 7;
    const int o = colTile * 16 + q;
    v8f acc = {};
    for (int kh = 0; kh < 3; ++kh) {
        for (int kw = 0; kw < 3; ++kw) {
            const int ih = 2 * oh - 1 + kh, iw = 2 * ow - 1 + kw;
            const bool valid = ((unsigned)ih < 16u) && ((unsigned)iw < 16u);
            const int kpos = kh * 3 + kw;
            size_t abase = 0;
            if (valid) abase = ((size_t)bimg * Ssz + ih * 16 + iw) * (size_t)Dsz;
            #pragma unroll 8
            for (int c0 = 0; c0 < Dsz; c0 += 4) {
                const int cc = c0 + 2 * half;
                v2f a, b;
                if (valid) {
                    float2 t = *(const float2*)(h + abase + cc);
                    a.x = t.x; a.y = t.y;
                } else { a.x = 0.f; a.y = 0.f; }
                b.x = cw[((size_t)o * Dsz + cc) * 9 + kpos];
                b.y = cw[((size_t)o * Dsz + cc + 1) * 9 + kpos];
                acc = wmma4(a, b, acc);
            }
        }
    }
    const int r0 = rowTile * 16 + 8 * half;
    #pragma unroll
    for (int j = 0; j < 8; ++j)
        x[(size_t)(r0 + j) * Dsz + o] = acc[j];
}

// ===================================================================
// 3) cnorm[k] = |codebooks[k]|^2 ; zero counts (must re-zero per call)
// ===================================================================
__global__ void cnorm_kernel(const float* __restrict__ cb,
                             float* __restrict__ cnorm,
                             int* __restrict__ counts) {
    const int k = blockIdx.x * 256 + threadIdx.x;         // 64 blocks
    const float4* row = (const float4*)(cb + (size_t)k * Dsz);
    float s = 0.f;
    #pragma unroll 8
    for (int i = 0; i < Dsz / 4; ++i) {
        float4 v = row[i];
        s += v.x * v.x + v.y * v.y + v.z * v.z + v.w * v.w;
    }
    cnorm[k] = s;
    counts[k] = 0;
}

// ===================================================================
// 4) argmin_k ( |c_k|^2 - 2 x.c_k )  — the dominant 34-GFLOP GEMM.
//    Block = 8 waves x 1 col slice; codebook tiles staged through LDS
//    (double-buffered, pad stride 132 for conflict-free b64 reads).
// ===================================================================
#define CB_STRIDE 132        // 128 + 4 pad: 16B-aligned rows, 64-bank spread

__global__ void dist_argmin(const float* __restrict__ x,
                            const float* __restrict__ cb,
                            const float* __restrict__ cnorm,
                            float* __restrict__ pV,
                            int* __restrict__ pI) {
    __shared__ float lds[2][16 * CB_STRIDE];              // 2 x 8.25KB

    const int slice    = blockIdx.x & 3;
    const int rowBlock = blockIdx.x >> 2;                 // 0..63
    const int waveId   = threadIdx.x >> 5;
    const int rowTile  = rowBlock * 8 + waveId;           // 0..511
    const int l = threadIdx.x & 31, half = l >> 4, q = l & 15;
    const int arow = rowTile * 16 + q;

    // staging role: thread tid -> codebook tile row r, cols c..c+7
    const int sr = threadIdx.x >> 4;                      // 0..15
    const int sc = (threadIdx.x & 15) * 8;                // 0,8,...,120

    // hoist A fragments (loop-invariant x row) into registers
    v2f afrag[32];
    #pragma unroll
    for (int k0 = 0; k0 < Dsz; k0 += 4)
        afrag[k0 >> 2] = *(const v2f*)(x + (size_t)arow * Dsz + k0 + 2 * half);

    float bestv[8];
    int   besti[8];
    #pragma unroll
    for (int j = 0; j < 8; ++j) { bestv[j] = 3.4e38f; besti[j] = 0; }

    const int colBase = slice * 4096;

    // prologue: stage tile 0 into buffer 0
    {
        const float4* src = (const float4*)(cb + (size_t)(colBase + sr) * Dsz + sc);
        float4 p0 = src[0], p1 = src[1];
        *(float4*)&lds[0][sr * CB_STRIDE + sc]     = p0;
        *(float4*)&lds[0][sr * CB_STRIDE + sc + 4] = p1;
    }
    __syncthreads();

    for (int t = 0; t < 256; ++t) {                       // 256 tiles of 16 cols
        const int buf = t & 1;
        const int col = colBase + t * 16 + q;

        // prefetch next tile into registers (overlaps with WMMAs below)
        float4 p0, p1;
        const bool more = (t + 1) < 256;
        if (more) {
            const float4* src = (const float4*)
                (cb + (size_t)(colBase + (t + 1) * 16 + sr) * Dsz + sc);
            p0 = src[0]; p1 = src[1];
        }

        v8f acc = {};
        #pragma unroll
        for (int k0 = 0; k0 < Dsz; k0 += 4) {             // 32 WMMAs
            const int kk = k0 + 2 * half;
            const v2f b = *(const v2f*)&lds[buf][q * CB_STRIDE + kk];
            acc = wmma4(afrag[k0 >> 2], b, acc);
        }

        const float cn = cnorm[col];
        #pragma unroll
        for (int j = 0; j < 8; ++j) {
            const float s = cn - 2.0f * acc[j];           // dist - |x|^2 (const/row)
            if (s < bestv[j]) { bestv[j] = s; besti[j] = col; }
        }

        if (more) {
            __syncthreads();                              // all done reading buf^1
            *(float4*)&lds[buf ^ 1][sr * CB_STRIDE + sc]     = p0;
            *(float4*)&lds[buf ^ 1][sr * CB_STRIDE + sc + 4] = p1;
            __syncthreads();
        }
    }

    // reduce over the 16 columns held in each half-wave
    const int r0 = rowTile * 16 + 8 * half;
    #pragma unroll
    for (int j = 0; j < 8; ++j) {
        float v = bestv[j]; int i = besti[j];
        #pragma unroll
        for (int m = 1; m < 16; m <<= 1) {
            const float ov = __shfl_xor(v, m, 32);
            const int   oi = __shfl_xor(i, m, 32);
            if (ov < v || (ov == v && oi < i)) { v = ov; i = oi; }
        }
        if (q == 0) {
            pV[(size_t)slice * Nrow + r0 + j] = v;
            pI[(size_t)slice * Nrow + r0 + j] = i;
        }
    }
}

// ===================================================================
// 5) Per-row finalize: combine slices, NSVQ noise substitution,
//    histogram counts, emit min_indices. One wave per row.
// ===================================================================
__global__ void finalize_rows(const float* __restrict__ x,
                              const float* __restrict__ cb,
                              const float* __restrict__ rnd,
                              const float* __restrict__ pV,
                              const int* __restrict__ pI,
                              float* __restrict__ quant,
                              int* __restrict__ counts,
                              float* __restrict__ outIdx) {
    const int n = blockIdx.x * 8 + (threadIdx.x >> 5);
    const int l = threadIdx.x & 31;
    float bv = pV[n]; int bi = pI[n];
    #pragma unroll
    for (int s = 1; s < 4; ++s) {
        const float v = pV[(size_t)s * Nrow + n];
        const int   i = pI[(size_t)s * Nrow + n];
        if (v < bv || (v == bv && i < bi)) { bv = v; bi = i; }
    }
    const float4 xv = ((const float4*)(x   + (size_t)n  * Dsz))[l];
    const float4 cq = ((const float4*)(cb  + (size_t)bi * Dsz))[l];
    const float4 rv = ((const float4*)(rnd + (size_t)n  * Dsz))[l];
    const float dx = xv.x - cq.x, dy = xv.y - cq.y,
                dz = xv.z - cq.z, dw = xv.w - cq.w;
    float sr = dx * dx + dy * dy + dz * dz + dw * dw;
    float sn = rv.x * rv.x + rv.y * rv.y + rv.z * rv.z + rv.w * rv.w;
    #pragma unroll
    for (int m = 1; m < 32; m <<= 1) {
        sr += __shfl_xor(sr, m, 32);
        sn += __shfl_xor(sn, m, 32);
    }
    const float scale = sqrtf(sr) / sqrtf(sn) + NEPS;
    float4 qv;
    qv.x = xv.x + scale * rv.x;  qv.y = xv.y + scale * rv.y;
    qv.z = xv.z + scale * rv.z;  qv.w = xv.w + scale * rv.w;
    ((float4*)(quant + (size_t)n * Dsz))[l] = qv;
    if (l == 0) {
        atomicAdd(&counts[bi], 1);
        outIdx[n] = (float)bi;
    }
}

// ===================================================================
// 6) perplexity + new_used
// ===================================================================
__global__ void stats_kernel(const int* __restrict__ counts,
                             const int* __restrict__ used,
                             float* __restrict__ outP,
                             float* __restrict__ outUsed) {
    __shared__ float red[256];
    float s = 0.f;
    for (int k = threadIdx.x; k < Ksz; k += 256) {
        const int c = counts[k];
        const float p = (float)c * (1.0f / (float)Nrow);
        s += -p * logf(p + NEPS);
        outUsed[k] = (float)(used[k] + c);
    }
    red[threadIdx.x] = s;
    __syncthreads();
    for (int ofs = 128; ofs > 0; ofs >>= 1) {
        if (threadIdx.x < ofs) red[threadIdx.x] += red[threadIdx.x + ofs];
        __syncthreads();
    }
    if (threadIdx.x == 0) outP[0] = expf(red[0]);
}

// ===================================================================
// 7) decode: out[b,j,:] = q[b,:,j] @ W_out + b_out  (raw reinterpret)
//    A[m=(b,j), k] = quant[b*8192 + k*64 + j]
// ===================================================================
__global__ void decode_gemm(const float* __restrict__ quant,
                            const float* __restrict__ Wout,   // [128,512]
                            const float* __restrict__ bout,
                            float* __restrict__ out) {        // [8192,512]
    const int w = blockIdx.x * 8 + (threadIdx.x >> 5);
    const int rowTile = w >> 5, colTile = w & 31;         // 512 x 32
    const int l = threadIdx.x & 31, half = l >> 4, q = l & 15;
    const int m = rowTile * 16 + q;
    const int bimg = m >> 6, jj = m & 63;
    const int col = colTile * 16 + q;
    v8f acc = {};
    #pragma unroll
    for (int k0 = 0; k0 < Dsz; k0 += 4) {
        const int kk = k0 + 2 * half;
        v2f a, b;
        a.x = quant[(size_t)bimg * 8192 + (size_t)kk * 64 + jj];
        a.y = quant[(size_t)bimg * 8192 + (size_t)(kk + 1) * 64 + jj];
        b.x = Wout[(size_t)kk * DIMsz + col];
        b.y = Wout[(size_t)(kk + 1) * DIMsz + col];
        acc = wmma4(a, b, acc);
    }
    const float bb = bout[col];
    const int r0 = rowTile * 16 + 8 * half;
    #pragma unroll
    for (int j = 0; j < 8; ++j)
        out[(size_t)(r0 + j) * DIMsz + col] = acc[j] + bb;
}

// ===================================================================
extern "C" void kernel_launch(void* const* d_in, const int* in_sizes, int n_in,
                              void* d_out, int out_size, void* d_ws, size_t ws_size,
                              hipStream_t stream) {
    const float* first = (const float*)d_in[0];
    const float* last  = (const float*)d_in[1];
    const float* rnd   = (const float*)d_in[2];
    const float* cb    = (const float*)d_in[3];
    const float* Win   = (const float*)d_in[4];
    const float* cw    = (const float*)d_in[6];
    const float* Wout  = (const float*)d_in[8];
    const float* bout  = (const float*)d_in[9];
    const int*   used  = (const int*)d_in[10];

    char* ws = (char*)d_ws;
    float* h     = (float*)(ws);                              // 16 MB
    float* x     = (float*)(ws + 16777216);                   //  4 MB
    float* quant = (float*)(ws + 20971520);                   //  4 MB
    float* cnorm = (float*)(ws + 25165824);                   // 64 KB
    int*   cnt   = (int*)  (ws + 25231360);                   // 64 KB
    float* pV    = (float*)(ws + 25296896);                   // 128 KB
    int*   pI    = (int*)  (ws + 25427968);                   // 128 KB

    float* out    = (float*)d_out;                            // [8192,512]
    float* outP   = out + 4194304;                            // perplexity
    float* outUsd = out + 4194305;                            // [16384]
    float* outIdx = out + 4194305 + Ksz;                      // [8192]

    encode_gemm<<<2048, 256, 0, stream>>>(first, last, Win, h);
    conv_gemm<<<512, 256, 0, stream>>>(h, cw, x);
    cnorm_kernel<<<64, 256, 0, stream>>>(cb, cnorm, cnt);
    dist_argmin<<<256, 256, 0, stream>>>(x, cb, cnorm, pV, pI);
    finalize_rows<<<1024, 256, 0, stream>>>(x, cb, rnd, pV, pI, quant, cnt, outIdx);
    stats_kernel<<<1, 256, 0, stream>>>(cnt, used, outP, outUsd);
    decode_gemm<<<2048, 256, 0, stream>>>(quant, Wout, bout, out);
}